// LogEig_52553219834313
// MI455X (gfx1250) — compile-verified
//
#include <hip/hip_runtime.h>

// LogEig via inverse scaling-and-squaring matrix logarithm.
// All heavy math is batched 64x64 f32 GEMM on the CDNA5 WMMA pipe
// (V_WMMA_F32_16X16X4_F32), one workgroup (8 wave32) per matrix, LDS-resident.

typedef __attribute__((ext_vector_type(2))) float v2f;
typedef __attribute__((ext_vector_type(8))) float v8f;

#define SQRT_STAGES 6   // k: number of repeated matrix square roots (2^k = 64)
#define NS_ITERS    12  // Newton-Schulz iterations per square root
#define LOG_TERMS   8   // Taylor terms for log(I+E)

// C(64x64) = alpha * A(64x64) * B(64x64) + diag_add * I
// A, B, C are LDS-resident row-major 64x64 f32.
// Executed cooperatively by 8 wave32 (256 threads); each wave computes two
// 16x16 output tiles, each tile = 16 WMMA f32 16x16x4 accumulations.
__device__ __forceinline__ void mm64(const float* As, const float* Bs, float* Cs,
                                     float alpha, float diag_add,
                                     int wave, int lane) {
  const int half = lane >> 4;   // lane-half: selects K-pair (A/B) and M+8 (C/D)
  const int ln   = lane & 15;
#pragma unroll
  for (int tt = 0; tt < 2; ++tt) {
    const int t  = wave + 8 * tt;
    const int ti = (t >> 2) << 4;   // output tile row base
    const int tj = (t & 3) << 4;    // output tile col base
    v8f acc = {0.f, 0.f, 0.f, 0.f, 0.f, 0.f, 0.f, 0.f};
#pragma unroll
    for (int kk = 0; kk < 16; ++kk) {
      const int k0 = (kk << 2) + (half << 1);
      // A fragment (16x4): lane row = ti+ln, a.x -> K=k0, a.y -> K=k0+1
      const v2f a = *reinterpret_cast<const v2f*>(&As[(ti + ln) * 64 + k0]);
      // B fragment (4x16): b.x -> row k0, b.y -> row k0+1, col = tj+ln
      v2f b;
      b.x = Bs[(k0 + 0) * 64 + tj + ln];
      b.y = Bs[(k0 + 1) * 64 + tj + ln];
      acc = __builtin_amdgcn_wmma_f32_16x16x4_f32(
          /*neg_a=*/false, a, /*neg_b=*/false, b,
          /*c_mod=*/(short)0, acc, /*reuse_a=*/false, /*reuse_b=*/false);
    }
#pragma unroll
    for (int r = 0; r < 8; ++r) {
      const int row = ti + r + (half << 3);
      const int col = tj + ln;
      float v = alpha * acc[r];
      v += (row == col) ? diag_add : 0.0f;
      Cs[row * 64 + col] = v;
    }
  }
}

__global__ __launch_bounds__(256) void logeig_kernel(const float* __restrict__ X,
                                                     float* __restrict__ Out) {
  // 4 x 16KB LDS matrix buffers = 64KB (WGP has a 320KB LDS pool).
  __shared__ float smem[4][4096];

  const int tid  = threadIdx.x;
  const int wave = tid >> 5;
  const int lane = tid & 31;

  const long long base = (long long)blockIdx.x * 4096;
  const float* Xb = X + base;
  float*       Ob = Out + base;

  // buffer roles (indices into smem, swapped by value)
  int iM = 0, iY = 1, iT = 2, iU = 3;

  // ---- load X (1024 float4 per matrix) ----
  for (int i = tid; i < 1024; i += 256)
    reinterpret_cast<float4*>(smem[iM])[i] =
        reinterpret_cast<const float4*>(Xb)[i];
  __syncthreads();

  // ---- trace-based spectral scaling: M = X/tr, log X = log(tr) I + log M ----
  float tr = 0.0f;
#pragma unroll
  for (int d = 0; d < 64; ++d) tr += smem[iM][d * 64 + d];  // LDS broadcast reads
  const float inv_s = 1.0f / tr;
  const float log_s = logf(tr);
  __syncthreads();  // all trace reads complete before scaling writes
  for (int i = tid; i < 4096; i += 256) smem[iM][i] *= inv_s;
  __syncthreads();

  // ---- k repeated matrix square roots via Newton-Schulz ----
  for (int stage = 0; stage < SQRT_STAGES; ++stage) {
    // Y0 = M  (Z0 = I implicitly; iterate M <- T M T, Y <- Y T, T = 1.5I - .5M)
    for (int i = tid; i < 4096; i += 256) smem[iY][i] = smem[iM][i];
    __syncthreads();
    for (int it = 0; it < NS_ITERS; ++it) {
      // T = 1.5 I - 0.5 M
      for (int i = tid; i < 4096; i += 256) {
        const int r = i >> 6, c = i & 63;
        smem[iT][i] = ((r == c) ? 1.5f : 0.0f) - 0.5f * smem[iM][i];
      }
      __syncthreads();
      mm64(smem[iT], smem[iM], smem[iU], 1.0f, 0.0f, wave, lane);  // U = T M
      __syncthreads();
      mm64(smem[iU], smem[iT], smem[iM], 1.0f, 0.0f, wave, lane);  // M = U T
      __syncthreads();
      mm64(smem[iY], smem[iT], smem[iU], 1.0f, 0.0f, wave, lane);  // U = Y T
      __syncthreads();
      int tmp = iY; iY = iU; iU = tmp;  // Y <- Y T
    }
    int tmp = iM; iM = iY; iY = tmp;    // M <- sqrt(M)
  }

  // ---- E = M - I ----
  for (int i = tid; i < 4096; i += 256) {
    const int r = i >> 6, c = i & 63;
    smem[iT][i] = smem[iM][i] - ((r == c) ? 1.0f : 0.0f);
  }
  __syncthreads();

  // ---- Horner for log(I+E) = E (c1 I + E (c2 I + ... )), c_j = (-1)^{j+1}/j ----
  {
    const float cm  = ((LOG_TERMS & 1) ? 1.0f : -1.0f) / (float)LOG_TERMS;
    const float cm1 = (((LOG_TERMS - 1) & 1) ? 1.0f : -1.0f) / (float)(LOG_TERMS - 1);
    for (int i = tid; i < 4096; i += 256) {
      const int r = i >> 6, c = i & 63;
      smem[iU][i] = cm * smem[iT][i] + ((r == c) ? cm1 : 0.0f);  // G = c_m E + c_{m-1} I
    }
    __syncthreads();
    for (int j = LOG_TERMS - 2; j >= 1; --j) {
      const float cj = ((j & 1) ? 1.0f : -1.0f) / (float)j;
      mm64(smem[iT], smem[iU], smem[iY], 1.0f, cj, wave, lane);  // Y = E G + c_j I
      __syncthreads();
      int tmp = iU; iU = iY; iY = tmp;                           // G <- Y
    }
  }

  // ---- log X = log(s) I + 2^k * (E G), fused into the final GEMM epilogue ----
  mm64(smem[iT], smem[iU], smem[iM], (float)(1 << SQRT_STAGES), log_s, wave, lane);
  __syncthreads();

  // ---- store ----
  for (int i = tid; i < 1024; i += 256)
    reinterpret_cast<float4*>(Ob)[i] = reinterpret_cast<float4*>(smem[iM])[i];
}

extern "C" void kernel_launch(void* const* d_in, const int* in_sizes, int n_in,
                              void* d_out, int out_size, void* d_ws, size_t ws_size,
                              hipStream_t stream) {
  const float* x = (const float*)d_in[0];
  float* out = (float*)d_out;
  const int B = in_sizes[0] / 4096;  // 8192 matrices of 64x64
  hipLaunchKernelGGL(logeig_kernel, dim3(B), dim3(256), 0, stream, x, out);
}